// Attention_31645319037130
// MI455X (gfx1250) — compile-verified
//
#include <hip/hip_runtime.h>
#include <math.h>

// Problem constants (from reference): B=32, S=8192, D=512, fp32 everywhere.
#define BATCH   32
#define SEQ     8192
#define DIM     512
#define SPLITS  64              // S-splits per batch (flash-decoding style)
#define SCHUNK  (SEQ / SPLITS)  // 128 keys per wave

#define NEG_INF_SCORE (-1e9f)

typedef __attribute__((ext_vector_type(2))) float v2f;
typedef __attribute__((ext_vector_type(8))) float v8f;

// ---------------------------------------------------------------------------
// Kernel 1: one wave32 per (batch, split). Online-softmax partial attention
// over SCHUNK keys. Scores computed with V_WMMA_F32_16X16X4_F32:
//   A (16x4 f32): row 0 = q[k0..k0+3], other rows 0
//   B (4x16 f32): column n = key[s0+n][k0..k0+3]   (16 keys per tile)
//   D row M=0    = partial scores, accumulated over all 128 k-steps.
// D row 0 lives in c8[0], lanes 0..15 (documented 16x16 f32 C/D layout).
// ---------------------------------------------------------------------------
__global__ __launch_bounds__(32)
void attn_partial_kernel(const float* __restrict__ q,
                         const float* __restrict__ key,
                         const float* __restrict__ val,
                         const int*   __restrict__ mask,
                         float* __restrict__ accbuf,   // [B*SPLITS, DIM]
                         float* __restrict__ mbuf,     // [B*SPLITS]
                         float* __restrict__ lbuf)     // [B*SPLITS]
{
    __shared__ __align__(16) float lds_q[DIM];
    __shared__ float lds_s[16];

    const int L     = threadIdx.x;        // lane 0..31
    const int split = blockIdx.x;
    const int b     = blockIdx.y;
    const int sbase = split * SCHUNK;

    // Stage q[b,:] into LDS once (it is reused 128x per chunk).
    for (int i = L; i < DIM; i += 32) lds_q[i] = q[b * DIM + i];
    __syncthreads();

    const int row  = L & 15;          // key row within 16-key tile / N index
    const int koff = (L >> 4) * 2;    // 0 or 2: which K-pair this lane holds

    const float* kbase = key + (size_t)b * SEQ * DIM;
    const float* vbase = val + (size_t)b * SEQ * DIM;
    const int*   mbase = mask + (size_t)b * SEQ;

    float m = -INFINITY;
    float l = 0.0f;
    // Lane L owns output dims {blk*128 + L*4 + c : blk<4, c<4} -> coalesced V.
    float acc[4][4];
    #pragma unroll
    for (int i = 0; i < 4; ++i)
        #pragma unroll
        for (int c = 0; c < 4; ++c) acc[i][c] = 0.0f;

    for (int ch = 0; ch < SCHUNK; ch += 16) {
        const int s0 = sbase + ch;

        // ---- score tile: 16 keys x full D via chained fp32 WMMA ----
        v8f c8 = {0.f, 0.f, 0.f, 0.f, 0.f, 0.f, 0.f, 0.f};
        const float* krow = kbase + (size_t)(s0 + row) * DIM + koff;
        const v2f zero2 = {0.f, 0.f};
        #pragma unroll 8
        for (int k0 = 0; k0 < DIM; k0 += 4) {
            v2f qv = *(const v2f*)(&lds_q[k0 + koff]);   // LDS broadcast
            v2f a  = (row == 0) ? qv : zero2;            // A row 0 only
            v2f bm = *(const v2f*)(krow + k0);           // coalesced b64 load
            c8 = __builtin_amdgcn_wmma_f32_16x16x4_f32(
                     false, a, false, bm, (short)0, c8, false, false);
        }

        // Scores for key n are in c8[0] of lane n (n = 0..15).
        float sc = c8[0];
        if (L < 16) {
            int mv = mbase[s0 + L];
            lds_s[L] = (mv == 0) ? NEG_INF_SCORE : sc;
        }
        __syncthreads();

        float sv[16];
        float cmax = -INFINITY;
        #pragma unroll
        for (int j = 0; j < 16; ++j) {
            sv[j] = lds_s[j];
            cmax = fmaxf(cmax, sv[j]);
        }
        __syncthreads();

        // ---- online softmax update ----
        const float mnew  = fmaxf(m, cmax);
        const float scale = __expf(m - mnew);   // 0 on first chunk (m=-inf)
        float p[16];
        float psum = 0.0f;
        #pragma unroll
        for (int j = 0; j < 16; ++j) {
            p[j] = __expf(sv[j] - mnew);
            psum += p[j];
        }
        l = l * scale + psum;
        #pragma unroll
        for (int i = 0; i < 4; ++i)
            #pragma unroll
            for (int c = 0; c < 4; ++c) acc[i][c] *= scale;

        // ---- acc += p * V : stream V once, float4 per lane per 128-dim blk ----
        #pragma unroll 4
        for (int j = 0; j < 16; ++j) {
            const float4* vrow = (const float4*)(vbase + (size_t)(s0 + j) * DIM);
            const float pj = p[j];
            #pragma unroll
            for (int i = 0; i < 4; ++i) {
                float4 v4 = vrow[i * 32 + L];
                acc[i][0] = fmaf(pj, v4.x, acc[i][0]);
                acc[i][1] = fmaf(pj, v4.y, acc[i][1]);
                acc[i][2] = fmaf(pj, v4.z, acc[i][2]);
                acc[i][3] = fmaf(pj, v4.w, acc[i][3]);
            }
        }
        m = mnew;
    }

    // ---- write partial (m, l, acc[DIM]) ----
    const size_t part = (size_t)b * SPLITS + split;
    float4* ap4 = (float4*)(accbuf + part * DIM);
    #pragma unroll
    for (int i = 0; i < 4; ++i) {
        float4 o;
        o.x = acc[i][0]; o.y = acc[i][1]; o.z = acc[i][2]; o.w = acc[i][3];
        ap4[i * 32 + L] = o;
    }
    if (L == 0) {
        mbuf[part] = m;
        lbuf[part] = l;
    }
}

// ---------------------------------------------------------------------------
// Kernel 2: combine the SPLITS partials per batch.
// out[b,d] = sum_s e^{m_s - M} acc_s[d] / sum_s l_s e^{m_s - M}
// ---------------------------------------------------------------------------
__global__ __launch_bounds__(128)
void attn_reduce_kernel(const float* __restrict__ accbuf,
                        const float* __restrict__ mbuf,
                        const float* __restrict__ lbuf,
                        float* __restrict__ out)
{
    const int b = blockIdx.x;
    const int t = threadIdx.x;           // 128 threads x float4 = 512 dims

    float M = -INFINITY;
    for (int s = 0; s < SPLITS; ++s)
        M = fmaxf(M, mbuf[b * SPLITS + s]);

    float Lsum = 0.0f;
    for (int s = 0; s < SPLITS; ++s)
        Lsum += lbuf[b * SPLITS + s] * __expf(mbuf[b * SPLITS + s] - M);

    float ox = 0.f, oy = 0.f, oz = 0.f, ow = 0.f;
    for (int s = 0; s < SPLITS; ++s) {
        const float w = __expf(mbuf[b * SPLITS + s] - M);
        const float4* ap =
            (const float4*)(accbuf + ((size_t)b * SPLITS + s) * DIM);
        float4 a4 = ap[t];
        ox = fmaf(w, a4.x, ox);
        oy = fmaf(w, a4.y, oy);
        oz = fmaf(w, a4.z, oz);
        ow = fmaf(w, a4.w, ow);
    }
    const float inv = 1.0f / Lsum;
    float4 o;
    o.x = ox * inv; o.y = oy * inv; o.z = oz * inv; o.w = ow * inv;
    ((float4*)(out + (size_t)b * DIM))[t] = o;
}

// ---------------------------------------------------------------------------
extern "C" void kernel_launch(void* const* d_in, const int* in_sizes, int n_in,
                              void* d_out, int out_size, void* d_ws, size_t ws_size,
                              hipStream_t stream) {
    const float* q    = (const float*)d_in[0];   // [B, D]
    const float* key  = (const float*)d_in[1];   // [B, S, D]
    const float* val  = (const float*)d_in[2];   // [B, S, D]
    const int*   mask = (const int*)d_in[3];     // [B, 1, S]
    float* out = (float*)d_out;                  // [B, D]

    // Workspace layout: acc partials, then m, then l.
    float* accbuf = (float*)d_ws;                         // B*SPLITS*DIM floats
    float* mbuf   = accbuf + (size_t)BATCH * SPLITS * DIM; // B*SPLITS
    float* lbuf   = mbuf + (size_t)BATCH * SPLITS;         // B*SPLITS

    dim3 grid1(SPLITS, BATCH, 1);
    attn_partial_kernel<<<grid1, 32, 0, stream>>>(q, key, val, mask,
                                                  accbuf, mbuf, lbuf);
    attn_reduce_kernel<<<BATCH, 128, 0, stream>>>(accbuf, mbuf, lbuf, out);
}